// MoEGate_19679540150990
// MI455X (gfx1250) — compile-verified
//
#include <hip/hip_runtime.h>
#include <math.h>

typedef __attribute__((ext_vector_type(2))) float v2f;
typedef __attribute__((ext_vector_type(8))) float v8f;

#define N_TOK 16384
#define DIM   2048
#define NEXP  64
#define TOPK  8
#define TOKB  32       // tokens per block (2 M-tiles)
#define KT    256      // K chunk staged in LDS
#define XPAD  4        // xs row stride 260 dwords -> conflict-free column reads
#define LPAD  8
#define NBLK  512      // 16384 / 32
#define BLK_PER_BATCH 128

// ---------------------------------------------------------------------------
// Main kernel: fp32 WMMA gate GEMM + softmax + top-8 + aux partials.
// Block = 128 threads (4 wave32). Wave w owns expert tile [16w,16w+16) and
// BOTH 16-token M-tiles: each B fragment load feeds two WMMAs.
// ---------------------------------------------------------------------------
__global__ __launch_bounds__(128) void moe_gate_main(
    const float* __restrict__ x, const float* __restrict__ wg,
    int* __restrict__ idx_out, float* __restrict__ w_out,
    float* __restrict__ probs_part, float* __restrict__ cnt_part)
{
  __shared__ float xs[TOKB][KT + XPAD];        // staged x tile (33 KB)
  __shared__ float lgs[TOKB][NEXP + LPAD];     // logits -> softmax scores
  __shared__ float scs[TOKB][NEXP + LPAD];     // mutable copy for top-k
  __shared__ int   cntl[NEXP];                 // block-local top-1 histogram

  const int tid  = threadIdx.x;                // 0..127
  const int lane = tid & 31;
  const int wave = tid >> 5;                   // 0..3 -> N tile
  const int n0   = wave * 16;
  const int tok0 = blockIdx.x * TOKB;

  if (tid < NEXP) cntl[tid] = 0;

  // WMMA fp32 16x16x4 A/B fragment striping (ISA §7.12.2):
  //   lanes 0-15: row=lane,  K pair {k,k+1} ; lanes 16-31: row=lane-16, {k+2,k+3}
  const int arow  = lane & 15;
  const int apair = (lane >> 4) * 2;

  v8f acc0 = {0.f,0.f,0.f,0.f,0.f,0.f,0.f,0.f};   // tokens [tok0,   tok0+16)
  v8f acc1 = {0.f,0.f,0.f,0.f,0.f,0.f,0.f,0.f};   // tokens [tok0+16,tok0+32)

  const int srow = tid >> 6;          // staging: 2 rows per iteration
  const int scol = (tid & 63) * 4;

  for (int k0 = 0; k0 < DIM; k0 += KT) {
    // Cooperative, coalesced global->LDS stage of the 32 x KT x-tile.
    #pragma unroll
    for (int it = 0; it < 16; ++it) {
      int r = 2 * it + srow;
      float4 v = *reinterpret_cast<const float4*>(
                     x + (size_t)(tok0 + r) * DIM + k0 + scol);
      *reinterpret_cast<float4*>(&xs[r][scol]) = v;
    }
    __syncthreads();

    // Prefetch next chunk of x (global_prefetch_b8) to overlap with WMMAs.
    if (k0 + KT < DIM) {
      #pragma unroll
      for (int it = 0; it < 16; ++it) {
        int r = 2 * it + srow;
        __builtin_prefetch(x + (size_t)(tok0 + r) * DIM + k0 + KT + scol, 0, 3);
      }
    }

    // B fragment: B[k][n] = W_gate[n][k]; one load feeds two WMMAs.
    const float* wrow = wg + (size_t)(n0 + arow) * DIM + k0 + apair;

    #pragma unroll 4
    for (int kk = 0; kk < KT; kk += 4) {
      v2f bfrag = *reinterpret_cast<const v2f*>(wrow + kk);                  // L2-hot
      v2f a0 = *reinterpret_cast<const v2f*>(&xs[arow][kk + apair]);         // ds b64
      v2f a1 = *reinterpret_cast<const v2f*>(&xs[16 + arow][kk + apair]);
      acc0 = __builtin_amdgcn_wmma_f32_16x16x4_f32(
          false, a0, false, bfrag, (short)0, acc0, false, false);
      acc1 = __builtin_amdgcn_wmma_f32_16x16x4_f32(
          false, a1, false, bfrag, (short)0, acc1, false, false);
    }
    __syncthreads();
  }

  // Scatter C tiles to LDS (ISA layout: VGPR j -> rows j and j+8).
  {
    const int crow = (lane >> 4) * 8;
    const int ccol = lane & 15;
    #pragma unroll
    for (int j = 0; j < 8; ++j) {
      lgs[crow + j     ][n0 + ccol] = acc0[j];
      lgs[crow + j + 16][n0 + ccol] = acc1[j];
    }
  }
  __syncthreads();

  // Per-token softmax + sorted top-8 (threads 0..31, one token each).
  if (tid < TOKB) {
    float mx = -3.402823e38f;
    for (int e = 0; e < NEXP; ++e) mx = fmaxf(mx, lgs[tid][e]);
    float sum = 0.f;
    for (int e = 0; e < NEXP; ++e) {
      float v = expf(lgs[tid][e] - mx);
      lgs[tid][e] = v; sum += v;
    }
    float inv = 1.f / sum;
    for (int e = 0; e < NEXP; ++e) {
      float v = lgs[tid][e] * inv;
      lgs[tid][e] = v;      // softmax scores kept for aux column-sums
      scs[tid][e] = v;      // mutable copy for selection
    }

    float tw[TOPK]; int ti[TOPK]; float wsum = 0.f;
    #pragma unroll
    for (int kk2 = 0; kk2 < TOPK; ++kk2) {       // repeated argmax => sorted,
      float best = -1.f; int bi = 0;             // first-index tie-break
      for (int e = 0; e < NEXP; ++e) {
        float v = scs[tid][e];
        if (v > best) { best = v; bi = e; }
      }
      ti[kk2] = bi; tw[kk2] = best; wsum += best;
      scs[tid][bi] = -2.f;
    }
    float invw = 1.f / (wsum + 1e-9f);
    int gtok = tok0 + tid;
    #pragma unroll
    for (int kk2 = 0; kk2 < TOPK; ++kk2) {
      idx_out[gtok * TOPK + kk2] = ti[kk2];
      w_out [gtok * TOPK + kk2] = tw[kk2] * invw;
    }
    atomicAdd(&cntl[ti[0]], 1);  // integer LDS atomic: order-independent
  }
  __syncthreads();

  // Deterministic per-block aux partials (no global float atomics).
  if (tid < NEXP) {
    float s = 0.f;
    for (int r = 0; r < TOKB; ++r) s += lgs[r][tid];
    probs_part[(size_t)blockIdx.x * NEXP + tid] = s;
    cnt_part [(size_t)blockIdx.x * NEXP + tid] = (float)cntl[tid];
  }
}

// ---------------------------------------------------------------------------
// Aux-loss reduction: fixed-order sums over the 128 block-partials per batch.
// ---------------------------------------------------------------------------
__global__ __launch_bounds__(256) void moe_gate_aux(
    const float* __restrict__ probs_part, const float* __restrict__ cnt_part,
    float* __restrict__ aux_out)
{
  __shared__ float probs_s[4][NEXP];
  __shared__ float cnt_s[4][NEXP];
  __shared__ float dot_s[4];

  const int t  = threadIdx.x;      // 0..255 -> (batch, expert)
  const int bb = t >> 6;
  const int e  = t & 63;

  float ps = 0.f, cs = 0.f;
  for (int blk = 0; blk < BLK_PER_BATCH; ++blk) {
    size_t o = ((size_t)(bb * BLK_PER_BATCH + blk)) * NEXP + e;
    ps += probs_part[o];
    cs += cnt_part[o];
  }
  probs_s[bb][e] = ps;
  cnt_s[bb][e]  = cs;
  __syncthreads();

  if (t < 4) {
    float psum = 0.f, csum = 0.f;
    for (int e2 = 0; e2 < NEXP; ++e2) { psum += probs_s[t][e2]; csum += cnt_s[t][e2]; }
    float pinv = 1.f / (psum + 1e-9f);
    float cinv = 1.f / (csum + 1e-9f);
    float dot = 0.f;
    for (int e2 = 0; e2 < NEXP; ++e2)
      dot += (probs_s[t][e2] * pinv) * (cnt_s[t][e2] * cinv);
    dot_s[t] = dot;
  }
  __syncthreads();

  if (t == 0) {
    float m = 0.f;
    for (int b2 = 0; b2 < 4; ++b2) m += dot_s[b2];
    aux_out[0] = (m * 0.25f) * ((float)NEXP * 0.01f);   // mean over b, * E * alpha
  }
}

// ---------------------------------------------------------------------------
extern "C" void kernel_launch(void* const* d_in, const int* in_sizes, int n_in,
                              void* d_out, int out_size, void* d_ws, size_t ws_size,
                              hipStream_t stream) {
  (void)in_sizes; (void)n_in; (void)out_size; (void)ws_size;

  const float* x  = (const float*)d_in[0];   // [4,4096,2048] f32
  const float* wg = (const float*)d_in[1];   // [64,2048]    f32

  float* out     = (float*)d_out;
  int*   idx_out = (int*)out;                             // 16384*8 int32
  float* w_out   = out + (size_t)N_TOK * TOPK;            // 16384*8 f32
  float* aux_out = out + (size_t)N_TOK * TOPK * 2;        // 1 f32

  float* probs_part = (float*)d_ws;                       // [512][64]
  float* cnt_part   = probs_part + (size_t)NBLK * NEXP;   // [512][64]

  moe_gate_main<<<NBLK, 128, 0, stream>>>(x, wg, idx_out, w_out,
                                          probs_part, cnt_part);
  moe_gate_aux<<<1, 256, 0, stream>>>(probs_part, cnt_part, aux_out);
}